// OptNetEq_39642548142386
// MI455X (gfx1250) — compile-verified
//
#include <hip/hip_runtime.h>
#include <math.h>

#define QPEN  0.1f
#define NVAR  100
#define NCON  50
#define BATCH 1024
#define ITERS 20
#define SIGMA 0.1f
#define STEPF 0.99f

#define MPAD    64    // constraint rows padded to 4 WMMA tiles
#define BSTRIDE 101   // odd stride -> conflict-free LDS fragment loads
#define SSTRIDE 65    // odd stride -> conflict-free column access in Cholesky

typedef __attribute__((ext_vector_type(2))) float v2f;
typedef __attribute__((ext_vector_type(8))) float v8f;

__global__ __launch_bounds__(256)
void qp_ipm_kernel(const float* __restrict__ x,
                   const float* __restrict__ A,
                   float* __restrict__ out)
{
    // weighted matrix B = A*diag(sqrt(w)), padded rows 50..63 stay zero
    __shared__ float sB[MPAD * BSTRIDE];
    // Schur complement S = B*B^T (then Cholesky factor in-place, lower tri)
    __shared__ float sS[MPAD * SSTRIDE];
    __shared__ float sp[NVAR], sz[NVAR], ss[NVAR], slam[NVAR];
    __shared__ float snu[MPAD];
    __shared__ float srp[NVAR], sr1[NVAR], srs[NVAR];
    __shared__ float sw[NVAR], ssw[NVAR], st1[NVAR];
    __shared__ float sdz[NVAR], sds[NVAR], sdlam[NVAR];
    __shared__ float srhs[MPAD];
    __shared__ float sred[128];
    __shared__ float sscal[2];   // [0]=mu, [1]=alpha

    const int tid  = threadIdx.x;
    const int bid  = blockIdx.x;
    const int lane = tid & 31;
    const int wid  = tid >> 5;

    __builtin_prefetch(A, 0, 3);           // global_prefetch_b8: pull A toward L2/L0

    // ---- init: zero B (incl. pad rows), load p, init IPM state ----
    for (int i = tid; i < MPAD * BSTRIDE; i += 256) sB[i] = 0.0f;
    if (tid < NVAR) {
        sp[tid]   = x[bid * NVAR + tid];
        sz[tid]   = 0.0f;
        ss[tid]   = 1.0f;
        slam[tid] = 1.0f;
    }
    if (tid < MPAD) snu[tid] = 0.0f;
    __syncthreads();

    for (int it = 0; it < ITERS; ++it) {
        // ---- residuals: rx, rp, s*lam partials ----
        if (tid < NVAR) {
            float atn = 0.0f;
            for (int j = 0; j < NCON; ++j) atn += A[j * NVAR + tid] * snu[j];
            float rx = QPEN * sz[tid] + sp[tid] - slam[tid] + atn;
            srp[tid]  = ss[tid] - sz[tid];
            sred[tid] = ss[tid] * slam[tid];
            sr1[tid]  = rx;                 // stash rx until mu is known
        }
        if (tid < NCON) {                   // ra = A z - 1
            float az = 0.0f;
            for (int k = 0; k < NVAR; ++k) az += A[tid * NVAR + k] * sz[k];
            srhs[tid] = az - 1.0f;
        }
        __syncthreads();

        if (tid == 0) {                     // mu = SIGMA * mean(s*lam)
            float acc = 0.0f;
            for (int i = 0; i < NVAR; ++i) acc += sred[i];
            sscal[0] = SIGMA * acc * (1.0f / NVAR);
        }
        __syncthreads();
        const float mu = sscal[0];

        if (tid < NVAR) {                   // rs, w = 1/(Q+lam/s), r1, t1 = w*r1
            float sv = ss[tid], lv = slam[tid];
            float rs = sv * lv - mu;
            srs[tid] = rs;
            float wv = sv / (QPEN * sv + lv);   // 1/(Q + lam/s)
            sw[tid]  = wv;
            ssw[tid] = sqrtf(wv);
            float r1 = -sr1[tid] + (lv * srp[tid] - rs) / sv;
            sr1[tid] = r1;
            st1[tid] = wv * r1;
        }
        __syncthreads();

        // ---- B = A * diag(sqrt(w)) into padded LDS ----
        for (int i = tid; i < NCON * NVAR; i += 256) {
            int r = i / NVAR, c = i - r * NVAR;
            sB[r * BSTRIDE + c] = A[i] * ssw[c];
        }
        __syncthreads();

        // ---- S = B * B^T via V_WMMA_F32_16X16X4_F32 (uniform, EXEC all-1) ----
        {
            const int j1 = wid & 3;         // column tile (shared by both tiles)
            const int i1 = wid >> 2;        // row tile #1
            const int i2 = i1 + 2;          // row tile #2
            const int mrow = lane & 15;
            const int koff = (lane >> 4) * 2;   // lanes 16-31 hold K=2,3
            const float* __restrict__ rb1 = &sB[(i1 * 16 + mrow) * BSTRIDE + koff];
            const float* __restrict__ rb2 = &sB[(i2 * 16 + mrow) * BSTRIDE + koff];
            const float* __restrict__ rbj = &sB[(j1 * 16 + mrow) * BSTRIDE + koff];
            v8f acc1 = {};
            v8f acc2 = {};
            #pragma unroll 5
            for (int k0 = 0; k0 < NVAR; k0 += 4) {
                v2f a1, a2, bb;
                a1.x = rb1[k0];  a1.y = rb1[k0 + 1];
                a2.x = rb2[k0];  a2.y = rb2[k0 + 1];
                bb.x = rbj[k0];  bb.y = rbj[k0 + 1];
                acc1 = __builtin_amdgcn_wmma_f32_16x16x4_f32(
                           false, a1, false, bb, (short)0, acc1, false, false);
                acc2 = __builtin_amdgcn_wmma_f32_16x16x4_f32(
                           false, a2, false, bb, (short)0, acc2, false, false);
            }
            const int ncol  = lane & 15;
            const int mhalf = (lane >> 4) * 8;  // lanes 16-31 hold M=8..15
            #pragma unroll
            for (int r = 0; r < 8; ++r) {
                sS[(i1 * 16 + mhalf + r) * SSTRIDE + j1 * 16 + ncol] = acc1[r];
                sS[(i2 * 16 + mhalf + r) * SSTRIDE + j1 * 16 + ncol] = acc2[r];
            }
        }
        __syncthreads();

        // ---- rhs = A*(w .* r1) + ra ----
        if (tid < NCON) {
            float acc = 0.0f;
            for (int k = 0; k < NVAR; ++k) acc += A[tid * NVAR + k] * st1[k];
            srhs[tid] += acc;
        }
        __syncthreads();

        // ---- Cholesky S = L L^T (in-place, lower triangle, 50x50 region) ----
        for (int k = 0; k < NCON; ++k) {
            if (tid == 0) sS[k * SSTRIDE + k] = sqrtf(sS[k * SSTRIDE + k]);
            __syncthreads();
            const float lkk = sS[k * SSTRIDE + k];
            if (tid > k && tid < NCON) sS[tid * SSTRIDE + k] /= lkk;
            __syncthreads();
            if (tid > k && tid < NCON) {
                const float lik = sS[tid * SSTRIDE + k];
                for (int j = k + 1; j <= tid; ++j)
                    sS[tid * SSTRIDE + j] -= lik * sS[j * SSTRIDE + k];
            }
            __syncthreads();
        }

        // ---- forward solve L y = rhs ----
        for (int k = 0; k < NCON; ++k) {
            if (tid == 0) srhs[k] /= sS[k * SSTRIDE + k];
            __syncthreads();
            const float yk = srhs[k];
            if (tid > k && tid < NCON) srhs[tid] -= sS[tid * SSTRIDE + k] * yk;
            __syncthreads();
        }
        // ---- backward solve L^T dnu = y (dnu left in srhs) ----
        for (int k = NCON - 1; k >= 0; --k) {
            if (tid == 0) srhs[k] /= sS[k * SSTRIDE + k];
            __syncthreads();
            const float xk = srhs[k];
            if (tid < k) srhs[tid] -= sS[k * SSTRIDE + tid] * xk;
            __syncthreads();
        }

        // ---- dz, ds, dlam, per-component step ratios ----
        if (tid < NVAR) {
            float atd = 0.0f;
            for (int j = 0; j < NCON; ++j) atd += A[j * NVAR + tid] * srhs[j];
            float dz   = sw[tid] * (sr1[tid] - atd);
            float dsv  = dz - srp[tid];
            float dlam = (-srs[tid] - slam[tid] * dsv) / ss[tid];
            sdz[tid]   = dz;
            sds[tid]   = dsv;
            sdlam[tid] = dlam;
            float r1m = (dsv  < 0.0f) ? (-ss[tid]   / dsv)  : 1e30f;
            float r2m = (dlam < 0.0f) ? (-slam[tid] / dlam) : 1e30f;
            sred[tid] = fminf(r1m, r2m);
        }
        __syncthreads();
        if (tid == 0) {                     // alpha = STEPF*min(1, min ratios)
            float mn = 1.0f;
            for (int i = 0; i < NVAR; ++i) mn = fminf(mn, sred[i]);
            sscal[1] = STEPF * mn;
        }
        __syncthreads();
        const float alpha = sscal[1];
        if (tid < NVAR) {
            sz[tid]   += alpha * sdz[tid];
            ss[tid]   += alpha * sds[tid];
            slam[tid] += alpha * sdlam[tid];
        }
        if (tid < NCON) snu[tid] += alpha * srhs[tid];
        __syncthreads();
    }

    if (tid < NVAR) out[bid * NVAR + tid] = sz[tid];
}

extern "C" void kernel_launch(void* const* d_in, const int* in_sizes, int n_in,
                              void* d_out, int out_size, void* d_ws, size_t ws_size,
                              hipStream_t stream) {
    const float* x = (const float*)d_in[0];   // (1024, 100) fp32
    const float* A = (const float*)d_in[1];   // (50, 100) fp32
    float* out = (float*)d_out;               // (1024, 100) fp32
    (void)in_sizes; (void)n_in; (void)out_size; (void)d_ws; (void)ws_size;
    hipLaunchKernelGGL(qp_ipm_kernel, dim3(BATCH), dim3(256), 0, stream, x, A, out);
}